// LongMaster_1056561955439
// MI455X (gfx1250) — compile-verified
//
#include <hip/hip_runtime.h>
#include <hip/hip_bf16.h>
#include <math.h>

// Shapes from the reference
#define S      2048
#define B      256
#define INSZ   32
#define HID    8
#define GATES  32          // 4*HID
#define OUTSZ  128
#define NBLK   80
#define SB     (S * B)     // 524288
#define OUT_ELEMS  ((size_t)SB * OUTSZ)   // 67108864
#define HF_OFF     OUT_ELEMS
#define CF_OFF     (OUT_ELEMS + (size_t)B * HID)

typedef __attribute__((ext_vector_type(2))) float v2f;
typedef __attribute__((ext_vector_type(8))) float v8f;

__device__ __forceinline__ float leaky(float v) {
    // v>=0 ? v : 0.01v  ==  max(v, 0.01v)
    return fmaxf(v, 0.01f * v);
}

// ---------------------------------------------------------------------------
// Kernel 1: LSTM scan, one wave32 per batch element.
// The x @ W_ih^T pre-gate GEMM is fused in via V_WMMA_F32_16X16X4_F32:
// each 16x16 WMMA M-tile = 16 consecutive timesteps of this wave's batch.
// Pre-gates for the chunk stay in accumulator VGPRs and feed the recurrence
// through ds_bpermute shuffles. Bias folded into accumulator init.
// ---------------------------------------------------------------------------
__global__ void __launch_bounds__(32)
lstm_scan_wmma(const float* __restrict__ x,      // [S,B,INSZ]
               const float* __restrict__ h0,     // [1,B,HID]
               const float* __restrict__ c0,     // [1,B,HID]
               const float* __restrict__ w_ih,   // [GATES,INSZ]
               const float* __restrict__ w_hh,   // [GATES,HID]
               const float* __restrict__ b_ih,   // [GATES]
               const float* __restrict__ b_hh,   // [GATES]
               float* __restrict__ ys,           // [S,B,HID] (workspace)
               float* __restrict__ hf,           // [B,HID]
               float* __restrict__ cf)           // [B,HID]
{
    const int j   = threadIdx.x & 31;   // lane == gate index
    const int b   = blockIdx.x;         // batch element
    const int lid = j & 15;
    const int hs  = j >> 4;             // lane half (A/B fragment K-pair select)

    // --- Preload B fragments: B = W_ih^T (K=INSZ x N=GATES), split in 2 N-halves.
    // ISA layout for 4x16 B tile (2 VGPRs): lane L, vgpr v -> K = 4t + 2*(L>>4) + v,
    // N = (L&15) (+16 for half 1). Element B[k][n] = w_ih[n*INSZ + k].
    v2f bfrag0[8], bfrag1[8];
#pragma unroll
    for (int t = 0; t < 8; ++t) {
        const int kb = 4 * t + 2 * hs;
        bfrag0[t].x = w_ih[(size_t)lid * INSZ + kb];
        bfrag0[t].y = w_ih[(size_t)lid * INSZ + kb + 1];
        bfrag1[t].x = w_ih[(size_t)(16 + lid) * INSZ + kb];
        bfrag1[t].y = w_ih[(size_t)(16 + lid) * INSZ + kb + 1];
    }
    const float bias0 = b_ih[lid] + b_hh[lid];
    const float bias1 = b_ih[16 + lid] + b_hh[16 + lid];

    // --- Recurrent weights: lane j holds row j of W_hh.
    float whh[HID];
#pragma unroll
    for (int k = 0; k < HID; ++k) whh[k] = w_hh[(size_t)j * HID + k];

    // --- State: lanes 0..7 own h_k, c_k.
    float hv = (j < HID) ? h0[(size_t)b * HID + j] : 0.0f;
    float cv = (j < HID) ? c0[(size_t)b * HID + j] : 0.0f;

    const int jg  = j & 7;
    const bool isG = (j >= 16) && (j < 24);     // g-gate lanes use tanh directly

    for (int s0 = 0; s0 < S; s0 += 16) {
        // A fragment rows: m = s0 + (L&15) timesteps of batch b, K-pair by lane half.
        const float* xa = x + ((size_t)(s0 + lid) * B + b) * INSZ + 2 * hs;

        v8f acc0, acc1;
#pragma unroll
        for (int v = 0; v < 8; ++v) { acc0[v] = bias0; acc1[v] = bias1; }

#pragma unroll
        for (int t = 0; t < 8; ++t) {
            v2f a = *(const v2f*)(xa + 4 * t);
            acc0 = __builtin_amdgcn_wmma_f32_16x16x4_f32(
                       false, a, false, bfrag0[t], (short)0, acc0, false, false);
            acc1 = __builtin_amdgcn_wmma_f32_16x16x4_f32(
                       false, a, false, bfrag1[t], (short)0, acc1, false, false);
        }

        // Consume the 16 timesteps sequentially.
        // acc layout: vgpr v, lane L -> s = s0 + v + 8*(L>>4), gate = (L&15) (+16 for acc1)
#pragma unroll
        for (int u = 0; u < 16; ++u) {
            const int soff = (u >= 8) ? 16 : 0;
            const int srcl = (j & 15) + soff;
            float g0 = __shfl(acc0[u & 7], srcl, 32);
            float g1 = __shfl(acc1[u & 7], srcl, 32);
            float g  = (j < 16) ? g0 : g1;   // lane j now holds pre-gate j

            // + h @ W_hh^T (broadcast h_k from lanes 0..7)
#pragma unroll
            for (int k = 0; k < HID; ++k)
                g = fmaf(__shfl(hv, k, 32), whh[k], g);

            // activations: sigmoid(x) = 0.5 + 0.5*tanh(x/2); g-gate lanes: tanh(x)
            float t  = tanhf(isG ? g : 0.5f * g);
            float a  = isG ? t : fmaf(0.5f, t, 0.5f);

            // gather i,f,g,o for hidden unit (j&7)
            float iv = __shfl(a, jg,      32);
            float fv = __shfl(a, jg + 8,  32);
            float tg = __shfl(a, jg + 16, 32);
            float ov = __shfl(a, jg + 24, 32);

            cv = fmaf(fv, cv, iv * tg);
            hv = ov * tanhf(cv);

            if (j < HID)
                ys[((size_t)(s0 + u) * B + b) * HID + j] = hv;
        }
    }

    if (j < HID) {
        hf[(size_t)b * HID + j] = hv;
        cf[(size_t)b * HID + j] = cv;
    }
}

// ---------------------------------------------------------------------------
// Kernel 2: head. One thread per (s,b) sample; weights staged in LDS.
// Final 1->128 layer is rank-1: block stages its 256 ybn scalars in LDS and
// emits the 128KB output slab with coalesced float4 stores.
// ---------------------------------------------------------------------------
__global__ void __launch_bounds__(256)
head_kernel(const float* __restrict__ ys,     // [SB, HID]
            const float* __restrict__ w_in,   // [4,8]
            const float* __restrict__ b_in,   // [4]
            const float* __restrict__ rb_w1,  // [NBLK,1,4]
            const float* __restrict__ rb_b1,  // [NBLK,1]
            const float* __restrict__ rb_w2,  // [NBLK,4,1]
            const float* __restrict__ rb_b2,  // [NBLK,4]
            const float* __restrict__ w_bn,   // [1,4]
            const float* __restrict__ b_bn,   // [1]
            const float* __restrict__ w_out,  // [128,1]
            const float* __restrict__ b_out,  // [128]
            float* __restrict__ out)          // [SB,128]
{
    __shared__ float s_win[32], s_bin[4];
    __shared__ float s_w1[NBLK * 4], s_b1[NBLK], s_w2[NBLK * 4], s_b2[NBLK * 4];
    __shared__ float s_wbn[4], s_bbn[1], s_wout[OUTSZ], s_bout[OUTSZ];
    __shared__ float s_ybn[256];

    const int tid = threadIdx.x;

    if (tid < 32)  s_win[tid] = w_in[tid];
    if (tid < 4)   s_bin[tid] = b_in[tid];
    if (tid < 4)   s_wbn[tid] = w_bn[tid];
    if (tid == 0)  s_bbn[0]   = b_bn[0];
    for (int i = tid; i < NBLK * 4; i += 256) {
        s_w1[i] = rb_w1[i];
        s_w2[i] = rb_w2[i];
        s_b2[i] = rb_b2[i];
    }
    if (tid < NBLK) s_b1[tid] = rb_b1[tid];
    if (tid < OUTSZ) { s_wout[tid] = w_out[tid]; s_bout[tid] = b_out[tid]; }
    __syncthreads();

    const size_t sample = (size_t)blockIdx.x * 256 + tid;

    float h[HID];
#pragma unroll
    for (int k = 0; k < HID; ++k) h[k] = ys[sample * HID + k];

    float y[4];
#pragma unroll
    for (int r = 0; r < 4; ++r) {
        float acc = s_bin[r];
#pragma unroll
        for (int k = 0; k < HID; ++k) acc = fmaf(h[k], s_win[r * HID + k], acc);
        y[r] = leaky(acc);
    }

    for (int blk = 0; blk < NBLK; ++blk) {
        float z = s_b1[blk];
#pragma unroll
        for (int r = 0; r < 4; ++r) z = fmaf(y[r], s_w1[blk * 4 + r], z);
        z = leaky(z);
#pragma unroll
        for (int r = 0; r < 4; ++r)
            y[r] = leaky(fmaf(z, s_w2[blk * 4 + r], y[r] + s_b2[blk * 4 + r]));
    }

    float ybn = s_bbn[0];
#pragma unroll
    for (int r = 0; r < 4; ++r) ybn = fmaf(y[r], s_wbn[r], ybn);
    ybn = leaky(ybn);

    s_ybn[tid] = ybn;
    __syncthreads();

    // Coalesced rank-1 output: block covers samples [blk*256, blk*256+256),
    // i.e. 256*128 floats written as float4.
    float* obase = out + (size_t)blockIdx.x * 256 * OUTSZ;
    for (int it = 0; it < 32; ++it) {
        const int idx4 = (it * 256 + tid) * 4;     // float index in [0, 32768)
        const int smp  = idx4 >> 7;                // local sample id
        const int o    = idx4 & 127;
        const float yb = s_ybn[smp];
        float4 v;
        v.x = 1.0f / (1.0f + __expf(-fmaf(yb, s_wout[o + 0], s_bout[o + 0])));
        v.y = 1.0f / (1.0f + __expf(-fmaf(yb, s_wout[o + 1], s_bout[o + 1])));
        v.z = 1.0f / (1.0f + __expf(-fmaf(yb, s_wout[o + 2], s_bout[o + 2])));
        v.w = 1.0f / (1.0f + __expf(-fmaf(yb, s_wout[o + 3], s_bout[o + 3])));
        *(float4*)(obase + idx4) = v;
    }
}

// ---------------------------------------------------------------------------
extern "C" void kernel_launch(void* const* d_in, const int* in_sizes, int n_in,
                              void* d_out, int out_size, void* d_ws, size_t ws_size,
                              hipStream_t stream) {
    const float* x     = (const float*)d_in[0];
    const float* h0    = (const float*)d_in[1];
    const float* c0    = (const float*)d_in[2];
    const float* w_ih  = (const float*)d_in[3];
    const float* w_hh  = (const float*)d_in[4];
    const float* b_ih  = (const float*)d_in[5];
    const float* b_hh  = (const float*)d_in[6];
    const float* w_in  = (const float*)d_in[7];
    const float* b_in  = (const float*)d_in[8];
    const float* rb_w1 = (const float*)d_in[9];
    const float* rb_b1 = (const float*)d_in[10];
    const float* rb_w2 = (const float*)d_in[11];
    const float* rb_b2 = (const float*)d_in[12];
    const float* w_bn  = (const float*)d_in[13];
    const float* b_bn  = (const float*)d_in[14];
    const float* w_out = (const float*)d_in[15];
    const float* b_out = (const float*)d_in[16];

    float* out = (float*)d_out;
    float* ys  = (float*)d_ws;              // SB*HID floats = 16 MB scratch

    // Phase 1: LSTM scan (one wave32 per batch element, WMMA-fused pre-gates)
    lstm_scan_wmma<<<B, 32, 0, stream>>>(x, h0, c0, w_ih, w_hh, b_ih, b_hh,
                                         ys, out + HF_OFF, out + CF_OFF);

    // Phase 2: head (one thread per sample, coalesced 256MB output)
    head_kernel<<<SB / 256, 256, 0, stream>>>(ys, w_in, b_in,
                                              rb_w1, rb_b1, rb_w2, rb_b2,
                                              w_bn, b_bn, w_out, b_out, out);
}